// GCNModel_22256520528145
// MI455X (gfx1250) — compile-verified
//
#include <hip/hip_runtime.h>
#include <hip/hip_bf16.h>

// ---------------------------------------------------------------------------
// GCN (3-layer GCNConv + LN + residual + mean-pool + MLP head) for gfx1250.
// f32 end-to-end (matches reference numerics); GEMMs via v_wmma_f32_16x16x4_f32.
// Weights packed K-pair-interleaved so every B fragment is one b64 load.
// ---------------------------------------------------------------------------

#define N_NODES   50000
#define N_EDGES   800000
#define HID       128
#define N_LAYERS  3
#define NUM_GRAPHS 500
#define LN_EPS    1e-5f

typedef __attribute__((ext_vector_type(2))) float v2f;
typedef __attribute__((ext_vector_type(8))) float v8f;

// ---------------------------------------------------------------------------
// Degree / normalization prep
// ---------------------------------------------------------------------------
__global__ void k_init_deg(float* deg, int n) {
    int i = blockIdx.x * blockDim.x + threadIdx.x;
    if (i < n) deg[i] = 1.0f;                     // self-loop
}

__global__ void k_count_deg(const int* __restrict__ dst, float* deg, int e) {
    int i = blockIdx.x * blockDim.x + threadIdx.x;
    if (i < e) atomicAdd(&deg[dst[i]], 1.0f);
}

__global__ void k_rsqrt_deg(float* dinv, int n) {
    int i = blockIdx.x * blockDim.x + threadIdx.x;
    if (i < n) dinv[i] = rsqrtf(dinv[i]);
}

__global__ void k_edge_norm(const int* __restrict__ src, const int* __restrict__ dst,
                            const float* __restrict__ dinv, float* __restrict__ enorm, int e) {
    int i = blockIdx.x * blockDim.x + threadIdx.x;
    if (i < e) enorm[i] = dinv[src[i]] * dinv[dst[i]];
}

// ---------------------------------------------------------------------------
// Pack transposed weights, K-pair interleaved:
//   Wp[m][kk][j] = float2{ W_m[j][2*kk], W_m[j][2*kk+1] },  kk in [0,64)
// so a B fragment (two adjacent K rows, same column) is one 8-byte load.
// m=0 -> W_in, m=1..3 -> Wc[l]
// ---------------------------------------------------------------------------
__global__ void k_pack_w(const float* __restrict__ W_in,
                         const float* __restrict__ Wc,
                         float2* __restrict__ Wp) {
    int idx = blockIdx.x * blockDim.x + threadIdx.x;   // 4 * 64 * 128 threads
    if (idx >= 4 * (HID / 2) * HID) return;
    int m  = idx / ((HID / 2) * HID);
    int kk = (idx / HID) % (HID / 2);
    int j  = idx % HID;
    const float* W = (m == 0) ? W_in : (Wc + (size_t)(m - 1) * HID * HID);
    float2 p;
    p.x = W[(size_t)j * HID + 2 * kk];
    p.y = W[(size_t)j * HID + 2 * kk + 1];
    Wp[idx] = p;
}

// ---------------------------------------------------------------------------
// WMMA GEMM: Out[n, :] = A[n, :] @ W^T (+ bias).  A:[nRows,128], Wp packed.
// 8 waves/block; each wave -> 16 rows x 128 cols (8 C fragments).
// K-loop step 4 using v_wmma_f32_16x16x4_f32 (full f32 precision).
// Per K-step: 1 x b64 A-load + 8 x b64 B-loads feed 8 WMMAs.
// ---------------------------------------------------------------------------
__global__ __launch_bounds__(256) void k_gemm128_wmma(
    const float* __restrict__ A, const float2* __restrict__ Wp,
    const float* __restrict__ bias, float* __restrict__ Out, int nRows) {

    const int lane    = threadIdx.x & 31;
    const int wave    = threadIdx.x >> 5;
    const int half    = lane >> 4;        // 0: lanes 0-15, 1: lanes 16-31
    const int l15     = lane & 15;
    const int nTiles  = nRows >> 4;       // 50000/16 = 3125, exact

    int rowTile = blockIdx.x * 8 + wave;
    const bool active = rowTile < nTiles;
    if (!active) rowTile = nTiles - 1;    // clamp: loads stay in-range, EXEC stays all-1s
    const int row0 = rowTile << 4;

    v8f c[8] = {};                        // 8 col-tiles of 16x16 f32 accumulators

    // A-frag layout (16x4 f32): lanes 0-15 hold (K=k0,k0+1), lanes 16-31 hold (K=k0+2,k0+3)
    const float* arow = A + (size_t)(row0 + l15) * HID;

    // Prefetch this wave's A strip (streams the 25.6MB activation matrix).
    __builtin_prefetch(arow, 0, 1);
    __builtin_prefetch(arow + HID / 2, 0, 1);

    for (int k0 = 0; k0 < HID; k0 += 4) {
        const int ka = k0 + 2 * half;                 // K base for this half-wave
        const v2f a = *(const v2f*)(arow + ka);       // one b64 load: A[M][ka], A[M][ka+1]
        // packed row index: kk = ka/2 = k0/2 + half
        const float2* wrow = Wp + (size_t)(k0 / 2 + half) * HID;
#pragma unroll
        for (int j = 0; j < 8; ++j) {
            const float2 bp = wrow[j * 16 + l15];     // one b64 load: B[ka][n], B[ka+1][n]
            v2f b; b.x = bp.x; b.y = bp.y;
            c[j] = __builtin_amdgcn_wmma_f32_16x16x4_f32(
                       false, a, false, b, (short)0, c[j], false, false);
        }
    }

    if (!active) return;
    // C/D layout: VGPR r -> row M=r (lanes 0-15) / M=r+8 (lanes 16-31), N=l15
#pragma unroll
    for (int j = 0; j < 8; ++j) {
        const int col = j * 16 + l15;
        const float bj = bias ? bias[col] : 0.0f;
#pragma unroll
        for (int r = 0; r < 8; ++r) {
            Out[(size_t)(row0 + r + 8 * half) * HID + col] = c[j][r] + bj;
        }
    }
}

// ---------------------------------------------------------------------------
// agg[n] = hw[n] * dinv[n]^2 + bc        (full overwrite: initializes agg)
// one wave per node, float4 per lane
// ---------------------------------------------------------------------------
__global__ void k_self_term(const float* __restrict__ hw, const float* __restrict__ dinv,
                            const float* __restrict__ bc, float* __restrict__ agg, int n) {
    int node = blockIdx.x * (blockDim.x >> 5) + (threadIdx.x >> 5);
    int lane = threadIdx.x & 31;
    if (node >= n) return;
    float sn = dinv[node]; sn *= sn;
    const float4 v = *(const float4*)(hw + (size_t)node * HID + lane * 4);
    const float4 b = *(const float4*)(bc + lane * 4);
    float4 o;
    o.x = v.x * sn + b.x; o.y = v.y * sn + b.y;
    o.z = v.z * sn + b.z; o.w = v.w * sn + b.w;
    *(float4*)(agg + (size_t)node * HID + lane * 4) = o;
}

// ---------------------------------------------------------------------------
// Edge scatter: agg[dst] += hw[src] * enorm[e].  One wave per edge.
// hw/agg are L2-resident (25.6MB each vs 192MB L2) -> atomics resolve in L2.
// ---------------------------------------------------------------------------
__global__ void k_edge_scatter(const float* __restrict__ hw, const float* __restrict__ enorm,
                               const int* __restrict__ src, const int* __restrict__ dst,
                               float* __restrict__ agg, int e) {
    int edge = blockIdx.x * (blockDim.x >> 5) + (threadIdx.x >> 5);
    int lane = threadIdx.x & 31;
    if (edge >= e) return;
    const float en = enorm[edge];
    const int s = src[edge], d = dst[edge];
    const float4 v = *(const float4*)(hw + (size_t)s * HID + lane * 4);
    float* out = agg + (size_t)d * HID + lane * 4;
    atomicAdd(out + 0, v.x * en);
    atomicAdd(out + 1, v.y * en);
    atomicAdd(out + 2, v.z * en);
    atomicAdd(out + 3, v.w * en);
}

// ---------------------------------------------------------------------------
// h = relu(LayerNorm(agg)*gamma + beta) + h     (wave per node, shfl reductions)
// ---------------------------------------------------------------------------
__global__ void k_ln_relu_res(const float* __restrict__ agg, float* __restrict__ h,
                              const float* __restrict__ gamma, const float* __restrict__ beta,
                              int n) {
    int node = blockIdx.x * (blockDim.x >> 5) + (threadIdx.x >> 5);
    int lane = threadIdx.x & 31;
    if (node >= n) return;
    const float4 v = *(const float4*)(agg + (size_t)node * HID + lane * 4);
    float s = v.x + v.y + v.z + v.w;
#pragma unroll
    for (int off = 16; off > 0; off >>= 1) s += __shfl_xor(s, off, 32);
    const float mu = s * (1.0f / HID);
    float4 d;
    d.x = v.x - mu; d.y = v.y - mu; d.z = v.z - mu; d.w = v.w - mu;
    float sq = d.x * d.x + d.y * d.y + d.z * d.z + d.w * d.w;
#pragma unroll
    for (int off = 16; off > 0; off >>= 1) sq += __shfl_xor(sq, off, 32);
    const float inv = rsqrtf(sq * (1.0f / HID) + LN_EPS);
    const float4 g = *(const float4*)(gamma + lane * 4);
    const float4 b = *(const float4*)(beta + lane * 4);
    float4 r = *(float4*)(h + (size_t)node * HID + lane * 4);
    r.x += fmaxf(d.x * inv * g.x + b.x, 0.0f);
    r.y += fmaxf(d.y * inv * g.y + b.y, 0.0f);
    r.z += fmaxf(d.z * inv * g.z + b.z, 0.0f);
    r.w += fmaxf(d.w * inv * g.w + b.w, 0.0f);
    *(float4*)(h + (size_t)node * HID + lane * 4) = r;
}

// ---------------------------------------------------------------------------
// Pooling
// ---------------------------------------------------------------------------
__global__ void k_zero(float* p, int n) {
    int i = blockIdx.x * blockDim.x + threadIdx.x;
    if (i < n) p[i] = 0.0f;
}

__global__ void k_pool_count(const int* __restrict__ batch, float* __restrict__ cnt, int n) {
    int i = blockIdx.x * blockDim.x + threadIdx.x;
    if (i < n) atomicAdd(&cnt[batch[i]], 1.0f);
}

__global__ void k_pool_sum(const float* __restrict__ h, const int* __restrict__ batch,
                           float* __restrict__ pooled, int n) {
    int node = blockIdx.x * (blockDim.x >> 5) + (threadIdx.x >> 5);
    int lane = threadIdx.x & 31;
    if (node >= n) return;
    const int g = batch[node];
    const float4 v = *(const float4*)(h + (size_t)node * HID + lane * 4);
    float* out = pooled + (size_t)g * HID + lane * 4;
    atomicAdd(out + 0, v.x);
    atomicAdd(out + 1, v.y);
    atomicAdd(out + 2, v.z);
    atomicAdd(out + 3, v.w);
}

// ---------------------------------------------------------------------------
// Head: out[g] = relu(mean_g @ W1^T + b1) @ W2^T + b2.  One 64-thread block/graph.
// ---------------------------------------------------------------------------
__global__ __launch_bounds__(64) void k_mlp(const float* __restrict__ pooled,
                                            const float* __restrict__ cnt,
                                            const float* __restrict__ W1, const float* __restrict__ b1,
                                            const float* __restrict__ W2, const float* __restrict__ b2,
                                            float* __restrict__ out) {
    const int g = blockIdx.x;
    const int j = threadIdx.x;                         // 64 hidden units
    const float inv = 1.0f / fmaxf(cnt[g], 1.0f);
    const float* p = pooled + (size_t)g * HID;
    const float* w = W1 + (size_t)j * HID;
    float acc = b1[j];
#pragma unroll 4
    for (int k = 0; k < HID; ++k) acc += (p[k] * inv) * w[k];
    __shared__ float sh[64];
    sh[j] = fmaxf(acc, 0.0f) * W2[j];
    __syncthreads();
    for (int s = 32; s > 0; s >>= 1) {
        if (j < s) sh[j] += sh[j + s];
        __syncthreads();
    }
    if (j == 0) out[g] = sh[0] + b2[0];
}

// ---------------------------------------------------------------------------
// Launch
// ---------------------------------------------------------------------------
extern "C" void kernel_launch(void* const* d_in, const int* in_sizes, int n_in,
                              void* d_out, int out_size, void* d_ws, size_t ws_size,
                              hipStream_t stream) {
    const float* x     = (const float*)d_in[0];
    const int*   ei    = (const int*)  d_in[1];
    const int*   batch = (const int*)  d_in[2];
    const float* W_in  = (const float*)d_in[3];
    const float* b_in  = (const float*)d_in[4];
    const float* Wc    = (const float*)d_in[5];
    const float* bc    = (const float*)d_in[6];
    const float* gamma = (const float*)d_in[7];
    const float* beta  = (const float*)d_in[8];
    const float* W1    = (const float*)d_in[9];
    const float* b1    = (const float*)d_in[10];
    const float* W2    = (const float*)d_in[11];
    const float* b2    = (const float*)d_in[12];
    float* out = (float*)d_out;

    const int* src = ei;
    const int* dst = ei + N_EDGES;

    // Workspace layout (floats, 256B-aligned offsets)
    float* ws     = (float*)d_ws;
    float* dinv   = ws;                    // 50000        -> [0, 50048)
    float* enorm  = ws + 50048;            // 800000       -> [50048, 850048)
    float2* Wp    = (float2*)(ws + 850048);// 4*64*128 float2 (= 4*128*128 floats)
    float* h      = ws + 915584;           // 50000*128
    float* hw     = ws + 7315584;          // 50000*128
    float* agg    = ws + 13715584;         // 50000*128
    float* pooled = ws + 20115584;         // 500*128
    float* cnt    = ws + 20179584;         // 500          (contiguous after pooled)

    const int B = 256;
    const int nWaveBlocksN = (N_NODES + 7) / 8;   // wave-per-node kernels, 8 waves/block
    const int nWaveBlocksE = (N_EDGES + 7) / 8;
    const int gemmBlocks   = ((N_NODES / 16) + 7) / 8;   // 3125 tiles / 8 waves

    // 1) degrees -> dinv, edge norms
    k_init_deg  <<<(N_NODES + B - 1) / B, B, 0, stream>>>(dinv, N_NODES);
    k_count_deg <<<(N_EDGES + B - 1) / B, B, 0, stream>>>(dst, dinv, N_EDGES);
    k_rsqrt_deg <<<(N_NODES + B - 1) / B, B, 0, stream>>>(dinv, N_NODES);
    k_edge_norm <<<(N_EDGES + B - 1) / B, B, 0, stream>>>(src, dst, dinv, enorm, N_EDGES);

    // 2) packed transposed weights
    k_pack_w<<<(4 * (HID / 2) * HID + B - 1) / B, B, 0, stream>>>(W_in, Wc, Wp);

    // 3) input projection: h = x @ W_in^T + b_in
    k_gemm128_wmma<<<gemmBlocks, B, 0, stream>>>(x, Wp, b_in, h, N_NODES);

    // 4) GCN layers
    for (int l = 0; l < N_LAYERS; ++l) {
        const float2* Wpl = Wp + (size_t)(1 + l) * (HID / 2) * HID;
        k_gemm128_wmma<<<gemmBlocks, B, 0, stream>>>(h, Wpl, nullptr, hw, N_NODES);
        k_self_term   <<<nWaveBlocksN, B, 0, stream>>>(hw, dinv, bc + (size_t)l * HID, agg, N_NODES);
        k_edge_scatter<<<nWaveBlocksE, B, 0, stream>>>(hw, enorm, src, dst, agg, N_EDGES);
        k_ln_relu_res <<<nWaveBlocksN, B, 0, stream>>>(agg, h,
                           gamma + (size_t)l * HID, beta + (size_t)l * HID, N_NODES);
    }

    // 5) global mean pool
    k_zero      <<<(NUM_GRAPHS * HID + NUM_GRAPHS + B - 1) / B, B, 0, stream>>>(pooled, NUM_GRAPHS * HID + NUM_GRAPHS);
    k_pool_count<<<(N_NODES + B - 1) / B, B, 0, stream>>>(batch, cnt, N_NODES);
    k_pool_sum  <<<nWaveBlocksN, B, 0, stream>>>(h, batch, pooled, N_NODES);

    // 6) MLP head
    k_mlp<<<NUM_GRAPHS, 64, 0, stream>>>(pooled, cnt, W1, b1, W2, b2, out);

    (void)in_sizes; (void)n_in; (void)out_size; (void)ws_size;
}